// gcn_air_75213467287803
// MI455X (gfx1250) — compile-verified
//
#include <hip/hip_runtime.h>
#include <math.h>

typedef __attribute__((ext_vector_type(2))) float v2f;
typedef __attribute__((ext_vector_type(8))) float v8f;

#define NF 128

// ---------------------------------------------------------------- zero
__global__ __launch_bounds__(256) void zero_f32(float* __restrict__ p, int n) {
  int i = blockIdx.x * 256 + threadIdx.x;
  int stride = gridDim.x * 256;
  for (; i < n; i += stride) p[i] = 0.0f;
}

// ---------------------------------------------------------------- SpMM (COO, wave per edge)
// lane handles features {lane, lane+32, lane+64, lane+96}: coalesced gather of
// src[col] (51MB matrix -> L2 resident) and 4 global_atomic_add_f32 per lane.
__global__ __launch_bounds__(256) void spmm_coo(
    const float* __restrict__ src, const int* __restrict__ erow,
    const int* __restrict__ ecol, const float* __restrict__ eval,
    float* __restrict__ dst, int E) {
  int wave = (blockIdx.x * 256 + threadIdx.x) >> 5;
  int lane = threadIdx.x & 31;
  if (wave >= E) return;
  int r = erow[wave];
  int c = ecol[wave];
  float v = eval[wave];
  const float* s = src + (size_t)c * NF;
  float* d = dst + (size_t)r * NF;
#pragma unroll
  for (int j = 0; j < 4; ++j) {
    int f = lane + j * 32;
    unsafeAtomicAdd(&d[f], v * s[f]);
  }
}

// ---------------------------------------------------------------- fp32 WMMA GEMM
// C[M x NCOLS] = A[M x 128] @ B[128 x NCOLS] (+bias).
// One 16x32 output block per wave (2 col tiles), K=128 -> 2x32 WMMA steps.
// NCOLS compile-time => all loop offsets are immediates (no per-step addr math).
// A layout (ISA 7.12.2, 32-bit A 16x4): lane holds A[lane%16][2*(lane>>4)+{0,1}]
// B layout (mirror):                    lane holds B[2*(lane>>4)+{0,1}][lane%16]
// C/D: VGPR r -> row (lane>=16 ? 8 : 0)+r, col lane%16.
template <int NCOLS, bool HAS_BIAS, bool HAS_C2>
__global__ __launch_bounds__(256) void gemm_wmma_f32(
    const float* __restrict__ A, const float* __restrict__ B,
    float* __restrict__ C, float* __restrict__ C2,
    const float* __restrict__ bias, int rowTiles) {
  constexpr int COLBLKS = NCOLS / 32;  // 16x32 blocks per row tile
  int wave = threadIdx.x >> 5;
  int lane = threadIdx.x & 31;
  int blk = blockIdx.x * 8 + wave;
  if (blk >= rowTiles * COLBLKS) return;
  int rt = blk / COLBLKS;           // compile-time divisor -> shifts
  int cb = blk - rt * COLBLKS;
  int laneLo = lane & 15;
  int laneHi = lane >> 4;
  int row = rt * 16 + laneLo;
  int col = cb * 32 + laneLo;

  // per-lane bases; inner loop uses only immediate offsets
  const float* arow = A + (size_t)row * NF + laneHi * 2;
  const float* b0 = B + (size_t)laneHi * 2 * NCOLS + col;
  const float* b1 = b0 + 16;

  v8f acc0 = {};
  v8f acc1 = {};
#pragma unroll
  for (int k0 = 0; k0 < NF; k0 += 4) {
    v2f a = *(const v2f*)(arow + k0);  // global_load_b64, 8B aligned
    v2f bx, by;
    bx.x = b0[k0 * NCOLS];
    bx.y = b0[k0 * NCOLS + NCOLS];
    by.x = b1[k0 * NCOLS];
    by.y = b1[k0 * NCOLS + NCOLS];
    acc0 = __builtin_amdgcn_wmma_f32_16x16x4_f32(false, a, false, bx, (short)0,
                                                 acc0, false, false);
    acc1 = __builtin_amdgcn_wmma_f32_16x16x4_f32(false, a, false, by, (short)0,
                                                 acc1, false, false);
  }

  float bv0 = 0.0f, bv1 = 0.0f;
  if (HAS_BIAS) {
    bv0 = bias[col];
    bv1 = bias[col + 16];
  }
  int rbase = rt * 16 + laneHi * 8;
  float* c0 = C + (size_t)rbase * NCOLS + col;
  float* d0 = HAS_C2 ? (C2 + (size_t)rbase * NCOLS + col) : nullptr;
#pragma unroll
  for (int r = 0; r < 8; ++r) {
    float v0 = acc0[r] + bv0;
    float v1 = acc1[r] + bv1;
    c0[r * NCOLS] = v0;
    c0[r * NCOLS + 16] = v1;
    if (HAS_C2) {
      d0[r * NCOLS] = v0;
      d0[r * NCOLS + 16] = v1;
    }
  }
}

// ---------------------------------------------------------------- attention gate + mix + PReLU
// wave per node: alpha = sigmoid([h, xin] . att_w + b); out = prelu((1-a)h + a*xin)
__global__ __launch_bounds__(256) void attmix_prelu(
    const float* __restrict__ h, const float* __restrict__ xin,
    const float* __restrict__ attw, const float* __restrict__ attb,
    const float* __restrict__ pw, float* __restrict__ out, int N) {
  int wave = (blockIdx.x * 256 + threadIdx.x) >> 5;
  int lane = threadIdx.x & 31;
  if (wave >= N) return;
  float4 hv = ((const float4*)(h + (size_t)wave * NF))[lane];
  float4 xv = ((const float4*)(xin + (size_t)wave * NF))[lane];
  float4 a1 = ((const float4*)attw)[lane];
  float4 a2 = ((const float4*)(attw + NF))[lane];
  float s = hv.x * a1.x + hv.y * a1.y + hv.z * a1.z + hv.w * a1.w +
            xv.x * a2.x + xv.y * a2.y + xv.z * a2.z + xv.w * a2.w;
#pragma unroll
  for (int m = 16; m > 0; m >>= 1) s += __shfl_xor(s, m, 32);
  float alpha = 1.0f / (1.0f + expf(-(s + attb[0])));
  float om = 1.0f - alpha;
  float p = pw[0];
  float4 o;
  o.x = om * hv.x + alpha * xv.x;
  o.y = om * hv.y + alpha * xv.y;
  o.z = om * hv.z + alpha * xv.z;
  o.w = om * hv.w + alpha * xv.w;
  o.x = o.x >= 0.0f ? o.x : p * o.x;
  o.y = o.y >= 0.0f ? o.y : p * o.y;
  o.z = o.z >= 0.0f ? o.z : p * o.z;
  o.w = o.w >= 0.0f ? o.w : p * o.w;
  ((float4*)(out + (size_t)wave * NF))[lane] = o;
}

// ---------------------------------------------------------------- elementwise PReLU
__global__ __launch_bounds__(256) void prelu_ew(
    const float* __restrict__ in, const float* __restrict__ pw,
    float* __restrict__ out, int n) {
  int i = blockIdx.x * 256 + threadIdx.x;
  int stride = gridDim.x * 256;
  float p = pw[0];
  for (; i < n; i += stride) {
    float v = in[i];
    out[i] = v >= 0.0f ? v : p * v;
  }
}

// ---------------------------------------------------------------- in-place log-softmax over 64 classes
__global__ __launch_bounds__(256) void log_softmax64(float* __restrict__ io, int N) {
  int wave = (blockIdx.x * 256 + threadIdx.x) >> 5;
  int lane = threadIdx.x & 31;
  if (wave >= N) return;
  float* row = io + (size_t)wave * 64;
  float v0 = row[lane];
  float v1 = row[lane + 32];
  float m = fmaxf(v0, v1);
#pragma unroll
  for (int d = 16; d > 0; d >>= 1) m = fmaxf(m, __shfl_xor(m, d, 32));
  float s = expf(v0 - m) + expf(v1 - m);
#pragma unroll
  for (int d = 16; d > 0; d >>= 1) s += __shfl_xor(s, d, 32);
  float l = m + logf(s);
  row[lane] = v0 - l;
  row[lane + 32] = v1 - l;
}

// ----------------------------------------------------------------
extern "C" void kernel_launch(void* const* d_in, const int* in_sizes, int n_in,
                              void* d_out, int out_size, void* d_ws, size_t ws_size,
                              hipStream_t stream) {
  (void)n_in; (void)out_size; (void)ws_size;
  const float* x    = (const float*)d_in[0];
  const int*   erow = (const int*)d_in[1];
  const int*   ecol = (const int*)d_in[2];
  const float* evl  = (const float*)d_in[3];
  const float* W0   = (const float*)d_in[4];
  const float* Wh   = (const float*)d_in[5];
  const float* attw = (const float*)d_in[6];
  const float* attb = (const float*)d_in[7];
  const float* outw = (const float*)d_in[8];
  const float* outb = (const float*)d_in[9];
  const float* pw   = (const float*)d_in[10];

  int N = in_sizes[0] / NF;   // 100000 (divisible by 16)
  int E = in_sizes[1];        // 1600000

  size_t per = (size_t)N * NF;
  float* ws  = (float*)d_ws;
  float* agg = ws;            // spmm accumulator
  float* h   = ws + per;      // current features
  float* xin = ws + 2 * per;  // residual input
  float* tmp = ws + 3 * per;  // mix/prelu output
  float* out = (float*)d_out;

  int rowTiles = N / 16;                 // 6250
  int spmmBlocks = (E + 7) / 8;
  int nodeBlocks = (N + 7) / 8;
  int g128 = (rowTiles * (NF / 32) + 7) / 8;   // 16x32 blocks, 8 waves/block
  int g64  = (rowTiles * (64 / 32) + 7) / 8;

  // hop 0: agg = A @ x ; h = xin = agg @ W0
  zero_f32<<<2048, 256, 0, stream>>>(agg, (int)per);
  spmm_coo<<<spmmBlocks, 256, 0, stream>>>(x, erow, ecol, evl, agg, E);
  gemm_wmma_f32<128, false, true><<<g128, 256, 0, stream>>>(
      agg, W0, h, xin, nullptr, rowTiles);

  // hops 1..2
  for (int hop = 0; hop < 2; ++hop) {
    attmix_prelu<<<nodeBlocks, 256, 0, stream>>>(h, xin, attw, attb, pw, tmp, N);
    zero_f32<<<2048, 256, 0, stream>>>(agg, (int)per);
    spmm_coo<<<spmmBlocks, 256, 0, stream>>>(tmp, erow, ecol, evl, agg, E);
    gemm_wmma_f32<128, false, false><<<g128, 256, 0, stream>>>(
        agg, Wh + (size_t)hop * NF * NF, h, nullptr, nullptr, rowTiles);
  }

  // head: logits = prelu(h) @ out_w + out_b ; log_softmax in place in d_out
  prelu_ew<<<2048, 256, 0, stream>>>(h, pw, tmp, (int)per);
  gemm_wmma_f32<64, true, false><<<g64, 256, 0, stream>>>(
      tmp, outw, out, nullptr, outb, rowTiles);
  log_softmax64<<<nodeBlocks, 256, 0, stream>>>(out, N);
}